// MGN_NET_5557687681768
// MI455X (gfx1250) — compile-verified
//
#include <hip/hip_runtime.h>

// MGN-NET NNConv stack on MI455X (gfx1250, wave32).
// Per-edge weight matrices W_e = relu(ea_e . nn_w + b) are built on the matrix
// pipes with V_WMMA_F32_16X16X4_F32 (fp32-exact), consumed immediately in
// registers, scatter-mean via global f32 atomics. Bias is folded into the GEMM
// via the spare K=6 slot (A[:,6]=1, packed-B row 6 = bias), C starts at inline
// zero, and weights+bias are packed into one array so every B load uses a
// uniform SGPR base with 32-bit scale_offset addressing (no per-lane 64-bit
// address math in the inner loop).

typedef __attribute__((ext_vector_type(2))) float v2f;
typedef __attribute__((ext_vector_type(8))) float v8f;

#define NN 35
#define NE (NN * NN)
#define NVF 6

// ---------------------------------------------------------------- zero init
__global__ void zero_ws_kernel(float* __restrict__ p, int n) {
  int i = blockIdx.x * blockDim.x + threadIdx.x;
  if (i < n) p[i] = 0.f;
}

// ---------------------------------------------------------------- weight pack
// pw[7][S]: rows 0..5 = nn_w rows, row 6 = nn_b (affine bias row).
__global__ void pack_w_kernel(const float* __restrict__ nw,
                              const float* __restrict__ nb,
                              float* __restrict__ pw, int S) {
  int idx = blockIdx.x * blockDim.x + threadIdx.x;
  if (idx >= 7 * S) return;
  pw[idx] = (idx < 6 * S) ? nw[idx] : nb[idx - 6 * S];
}

// ---------------------------------------------------------------- layer 1 (cin = 1)
__global__ void layer1_kernel(const float* __restrict__ x,
                              const float* __restrict__ ea,
                              const float* __restrict__ nw,
                              const float* __restrict__ nb,
                              const float* __restrict__ root,
                              const float* __restrict__ bias,
                              float* __restrict__ h1) {
  int idx = blockIdx.x * blockDim.x + threadIdx.x;  // 35*256
  if (idx >= NN * 256) return;
  int j = idx >> 8, o = idx & 255;
  float w0 = nw[0 * 256 + o], w1 = nw[1 * 256 + o], w2 = nw[2 * 256 + o];
  float w3 = nw[3 * 256 + o], w4 = nw[4 * 256 + o], w5 = nw[5 * 256 + o];
  float b = nb[o];
  float acc = 0.f;
  for (int i = 0; i < NN; ++i) {
    const float* e6 = ea + (i * NN + j) * NVF;
    float w = b + e6[0] * w0 + e6[1] * w1 + e6[2] * w2 +
                  e6[3] * w3 + e6[4] * w4 + e6[5] * w5;
    acc += x[i] * fmaxf(w, 0.f);
  }
  float val = acc * (1.f / NN) + x[j] * root[o] + bias[o];
  h1[idx] = fmaxf(val, 0.f);
}

// ---------------------------------------------------------------- WMMA message pass
// One wave per (src i, 16-wide j group, 32-wide o strip = 2 o-tiles).
// Per channel c: W[16j][16o] = relu(A x B) with bias in the K=6 slot,
// then msg += h[i,c] * W (wave-uniform SGPR pk_fma).
__global__ void __launch_bounds__(32)
nnconv_msg_wmma(const float* __restrict__ ea,    // [NE][6]
                const float* __restrict__ pw,    // [7][cin*cout] packed w+bias
                const float* __restrict__ hsrc,  // [NN][cin]
                float* __restrict__ agg,         // [NN][cout] pre-zeroed
                int cin, int cout) {
  const int lane = threadIdx.x;      // one wave32
  const int i    = blockIdx.x;       // source node
  const int j0   = blockIdx.y * 16;  // dest group base
  const int o0   = blockIdx.z * 32;  // 32-wide output strip
  const int half = lane >> 4;
  const int m    = lane & 15;        // A row (j) and B/D column (o)
  const int S    = cin * cout;       // pw row stride

  // A fragments (c-invariant). 16x4 f32 A layout: VGPR r, K = r + 2*half.
  const int  j  = j0 + m;
  const bool jv = (j < NN);
  const float* e6 = ea + (i * NN + (jv ? j : 0)) * NVF;
  v2f a1, a2;
  if (half == 0) {
    a1.x = jv ? e6[0] : 0.f;  a1.y = jv ? e6[1] : 0.f;   // K=0,1
    a2.x = jv ? e6[4] : 0.f;  a2.y = jv ? e6[5] : 0.f;   // K=4,5
  } else {
    a1.x = jv ? e6[2] : 0.f;  a1.y = jv ? e6[3] : 0.f;   // K=2,3
    a2.x = jv ? 1.0f  : 0.f;  a2.y = 0.f;                // K=6 affine-1, K=7 pad
  }

  // Lane-constant 32-bit B indices against the single uniform base `pw`.
  // B 4x16 layout: VGPR r holds row K = base + r + 2*half.
  // K=6 -> bias row; K=7 aliases row 6 (A column K=7 is all-zero).
  const int base = o0 + m;
  const int R1x  = (half ? 2 : 0) * S + base;
  const int R1y  = (half ? 3 : 1) * S + base;
  const int R2x  = (half ? 6 : 4) * S + base;
  const int R2y  = (half ? 6 : 5) * S + base;

  v8f msg0 = {0.f, 0.f, 0.f, 0.f, 0.f, 0.f, 0.f, 0.f};
  v8f msg1 = {0.f, 0.f, 0.f, 0.f, 0.f, 0.f, 0.f, 0.f};
  const float* hrow = hsrc + i * cin;

  for (int c = 0; c < cin; ++c) {
    const int off = c * cout;

    v2f b1a, b2a, b1b, b2b;
    b1a.x = pw[R1x + off];       b1a.y = pw[R1y + off];
    b2a.x = pw[R2x + off];       b2a.y = pw[R2y + off];
    b1b.x = pw[R1x + off + 16];  b1b.y = pw[R1y + off + 16];
    b2b.x = pw[R2x + off + 16];  b2b.y = pw[R2y + off + 16];

    v8f w0 = {0.f, 0.f, 0.f, 0.f, 0.f, 0.f, 0.f, 0.f};
    v8f w1 = {0.f, 0.f, 0.f, 0.f, 0.f, 0.f, 0.f, 0.f};
    w0 = __builtin_amdgcn_wmma_f32_16x16x4_f32(false, a1, false, b1a,
                                               (short)0, w0, false, false);
    w0 = __builtin_amdgcn_wmma_f32_16x16x4_f32(false, a2, false, b2a,
                                               (short)0, w0, false, false);
    w1 = __builtin_amdgcn_wmma_f32_16x16x4_f32(false, a1, false, b1b,
                                               (short)0, w1, false, false);
    w1 = __builtin_amdgcn_wmma_f32_16x16x4_f32(false, a2, false, b2b,
                                               (short)0, w1, false, false);

    const float hc = hrow[c];  // wave-uniform -> SGPR
#pragma unroll
    for (int r = 0; r < 8; ++r) {
      msg0[r] += hc * fmaxf(w0[r], 0.f);
      msg1[r] += hc * fmaxf(w1[r], 0.f);
    }
  }

  // D layout: VGPR r, lane l -> row M = r + 8*half, col N = lane&15.
#pragma unroll
  for (int r = 0; r < 8; ++r) {
    int jj = j0 + r + 8 * half;
    if (jj < NN) {
      atomicAdd(&agg[jj * cout + o0 + m],      msg0[r]);
      atomicAdd(&agg[jj * cout + o0 + 16 + m], msg1[r]);
    }
  }
}

// ---------------------------------------------------------------- root + bias + relu
__global__ void combine_kernel(const float* __restrict__ agg,
                               const float* __restrict__ hprev,
                               const float* __restrict__ root,
                               const float* __restrict__ bias,
                               float* __restrict__ hout,
                               int cin, int cout) {
  int idx = blockIdx.x * blockDim.x + threadIdx.x;
  if (idx >= NN * cout) return;
  int j = idx / cout, o = idx % cout;
  float acc = agg[idx] * (1.f / NN);
  const float* hp = hprev + j * cin;
  for (int c = 0; c < cin; ++c) acc += hp[c] * root[c * cout + o];
  acc += bias[o];
  hout[idx] = fmaxf(acc, 0.f);
}

// ---------------------------------------------------------------- cbt output
__global__ void cbt_kernel(const float* __restrict__ h, float* __restrict__ out) {
  int idx = blockIdx.x * blockDim.x + threadIdx.x;
  if (idx >= NE) return;
  int i = idx / NN, j = idx % NN;
  const float* hi = h + i * 128;
  const float* hj = h + j * 128;
  float s = 0.f;
  for (int f = 0; f < 128; ++f) s += fabsf(hj[f] - hi[f]);
  out[idx] = s;
}

// ---------------------------------------------------------------- launch
extern "C" void kernel_launch(void* const* d_in, const int* in_sizes, int n_in,
                              void* d_out, int out_size, void* d_ws, size_t ws_size,
                              hipStream_t stream) {
  const float* x     = (const float*)d_in[0];
  const float* ea    = (const float*)d_in[1];
  // d_in[2] = edge_index: dense graph, e = i*35 + j (src=i, dst=j), hardcoded.
  const float* nn1w  = (const float*)d_in[3];
  const float* nn1b  = (const float*)d_in[4];
  const float* nn2w  = (const float*)d_in[5];
  const float* nn2b  = (const float*)d_in[6];
  const float* nn3w  = (const float*)d_in[7];
  const float* nn3b  = (const float*)d_in[8];
  const float* root1 = (const float*)d_in[9];
  const float* bias1 = (const float*)d_in[10];
  const float* root2 = (const float*)d_in[11];
  const float* bias2 = (const float*)d_in[12];
  const float* root3 = (const float*)d_in[13];
  const float* bias3 = (const float*)d_in[14];

  const int S2 = 256 * 256;  // layer-2 nn_w row stride
  const int S3 = 256 * 128;  // layer-3 nn_w row stride

  float* ws   = (float*)d_ws;
  float* h1   = ws;               // 35*256
  float* h2   = h1 + NN * 256;    // 35*256
  float* h3   = h2 + NN * 256;    // 35*128
  float* agg2 = h3 + NN * 128;    // 35*256 \ contiguous -> single zero pass
  float* agg3 = agg2 + NN * 256;  // 35*128 /
  float* pw2  = agg3 + NN * 128;  // 7*S2 packed weights+bias, layer 2
  float* pw3  = pw2 + 7 * S2;     // 7*S3 packed weights+bias, layer 3

  const int nzero = NN * 256 + NN * 128;
  zero_ws_kernel<<<(nzero + 255) / 256, 256, 0, stream>>>(agg2, nzero);
  pack_w_kernel<<<(7 * S2 + 255) / 256, 256, 0, stream>>>(nn2w, nn2b, pw2, S2);
  pack_w_kernel<<<(7 * S3 + 255) / 256, 256, 0, stream>>>(nn3w, nn3b, pw3, S3);

  layer1_kernel<<<(NN * 256 + 255) / 256, 256, 0, stream>>>(
      x, ea, nn1w, nn1b, root1, bias1, h1);

  // Layer 2: cin=256, cout=256. grid = (35 src) x (3 j-groups) x (8 o-strips).
  nnconv_msg_wmma<<<dim3(NN, 3, 8), 32, 0, stream>>>(
      ea, pw2, h1, agg2, 256, 256);
  combine_kernel<<<(NN * 256 + 255) / 256, 256, 0, stream>>>(
      agg2, h1, root2, bias2, h2, 256, 256);

  // Layer 3: cin=256, cout=128. grid = (35) x (3) x (4 o-strips).
  nnconv_msg_wmma<<<dim3(NN, 3, 4), 32, 0, stream>>>(
      ea, pw3, h2, agg3, 256, 128);
  combine_kernel<<<(NN * 128 + 255) / 256, 256, 0, stream>>>(
      agg3, h2, root3, bias3, h3, 256, 128);

  cbt_kernel<<<(NE + 255) / 256, 256, 0, stream>>>(h3, (float*)d_out);
}